// EuclideanCodebook_81767587381979
// MI455X (gfx1250) — compile-verified
//
#include <hip/hip_runtime.h>
#include <hip/hip_bf16.h>
#include <stdint.h>

typedef __attribute__((ext_vector_type(2))) float v2f;
typedef __attribute__((ext_vector_type(4))) float v4f;
typedef __attribute__((ext_vector_type(8))) float v8f;
typedef __attribute__((ext_vector_type(4))) int   v4i;

#define DIM     128
#define KCODES  1024
#define NROWS   (32 * 4096)
#define SX      132                 // padded LDS row stride (floats), conflict-free frag loads
#define BM      128                 // rows per block (one 16-row band per wave)
#define BN      64                  // codebook columns per chunk (4 N-tiles per wave)
#define NCHUNK  (KCODES / BN)       // 16
#define ESBUF   (BN * SX)           // floats per embed LDS buffer

__device__ __forceinline__ float qt7(float v) {
    return rintf(v * 1e7f) * 1e-7f; // jnp.round(v*1e7)/1e7, RNE
}

// ---- CDNA5 async global->LDS (ASYNCcnt) with portable fallback -------------
// Probe-derived signature: (as1 v4i*, as3 v4i*, imm offset, imm cpol)
__device__ __forceinline__ void async_b128(const float* gsrc, float* ldst) {
#if __has_builtin(__builtin_amdgcn_global_load_async_to_lds_b128)
    typedef __attribute__((address_space(1))) v4i gv4i_t;
    typedef __attribute__((address_space(3))) v4i lv4i_t;
    __builtin_amdgcn_global_load_async_to_lds_b128(
        (gv4i_t*)(uintptr_t)gsrc,
        (lv4i_t*)(uint32_t)(uintptr_t)ldst, 0, 0);
#else
    *(v4f*)ldst = *(const v4f*)gsrc;
#endif
}
__device__ __forceinline__ void wait_async0() {
#if __has_builtin(__builtin_amdgcn_global_load_async_to_lds_b128)
#if __has_builtin(__builtin_amdgcn_s_wait_asynccnt)
    __builtin_amdgcn_s_wait_asynccnt(0);
#else
    asm volatile("s_wait_asynccnt 0x0" ::: "memory");
#endif
#endif
}

// ---------------------------------------------------------------- zero ws
__global__ void vq_zero(float* __restrict__ p, int n) {
    int i = blockIdx.x * 256 + threadIdx.x;
    if (i < n) p[i] = 0.0f;
}

// ---------------------------------------------------------------- e_sq[k] = qt(sum(qt(embed[k])^2))
__global__ __launch_bounds__(256)
void vq_esq(const float* __restrict__ embed, float* __restrict__ e_sq) {
    int gid  = blockIdx.x * 256 + threadIdx.x;
    int k    = gid >> 5;
    int lane = threadIdx.x & 31;
    if (k >= KCODES) return;
    v4f v = *(const v4f*)(embed + (size_t)k * DIM + lane * 4);
    float a = qt7(v.x), b = qt7(v.y), c = qt7(v.z), d = qt7(v.w);
    float s = a * a + b * b + c * c + d * d;
    #pragma unroll
    for (int m = 16; m >= 1; m >>= 1) s += __shfl_xor(s, m, 32);
    if (lane == 0) e_sq[k] = qt7(s);
}

// ---------------------------------------------------------------- main kernel
__global__ __launch_bounds__(256)
void vq_main(const float* __restrict__ x, const float* __restrict__ embed,
             const float* __restrict__ e_sq,
             float* __restrict__ out_quant, float* __restrict__ out_ind,
             float* __restrict__ ws_counts, float* __restrict__ ws_sums)
{
    __shared__ float xs[BM * SX];       // 67,584 B : x tile 128 x 128
    __shared__ float es[2 * ESBUF];     // 67,584 B : double-buffered embed tile 64 x 128
    __shared__ float eqs[KCODES];       //  4,096 B : e_sq table
    __shared__ int   finalIdx[BM];

    const int t    = threadIdx.x;
    const int lane = t & 31;
    const int w    = t >> 5;            // wave 0..7 -> owns rows w*16..w*16+15
    const int rowBlock = blockIdx.x * BM;

    // f32 WMMA fragment layout (ISA 7.12.2): lanes 0-15 K0/K1, lanes 16-31 K2/K3
    const int mrow = lane & 15;
    const int ksel = (lane >> 4) << 1;
    const int hi   = lane >> 4;

    // ---- prologue: async-stage x tile, e_sq, and embed chunk 0 ----
    for (int s = t; s < BM * (DIM / 4); s += 256) {
        int r = s >> 5, d4 = (s & 31) << 2;
        async_b128(x + (size_t)(rowBlock + r) * DIM + d4, &xs[r * SX + d4]);
    }
    async_b128(e_sq + t * 4, &eqs[t * 4]);
    for (int s = t; s < BN * (DIM / 4); s += 256) {
        int r = s >> 5, d4 = (s & 31) << 2;
        async_b128(embed + (size_t)r * DIM + d4, &es[r * SX + d4]);
    }
    wait_async0();
    __syncthreads();

    float bestVal[8];
    int   bestIdx[8];
    #pragma unroll
    for (int r = 0; r < 8; ++r) { bestVal[r] = -3.4e38f; bestIdx[r] = 0; }

    for (int c = 0; c < NCHUNK; ++c) {
        // issue async copy of next embed chunk into the other buffer (overlaps WMMA)
        if (c + 1 < NCHUNK) {
            float* esn = es + ((c + 1) & 1) * ESBUF;
            const float* gsrc = embed + (size_t)(c + 1) * BN * DIM;
            for (int s = t; s < BN * (DIM / 4); s += 256) {
                int r = s >> 5, d4 = (s & 31) << 2;
                async_b128(gsrc + (size_t)r * DIM + d4, &esn[r * SX + d4]);
            }
        }

        // 16 rows x 64 cols via 4 accumulators sharing the A fragment:
        // 5 ds_b64 loads per 4 v_wmma_f32_16x16x4_f32
        const float* esb = es + (c & 1) * ESBUF;
        v8f acc0 = {}, acc1 = {}, acc2 = {}, acc3 = {};
        const float* ap = &xs[(w * 16 + mrow) * SX + ksel];
        const float* bp = &esb[mrow * SX + ksel];
        #pragma unroll
        for (int d0 = 0; d0 < DIM; d0 += 4) {
            v2f a  = *(const v2f*)(ap + d0);
            v2f b0 = *(const v2f*)(bp + d0);
            v2f b1 = *(const v2f*)(bp + 16 * SX + d0);
            v2f b2 = *(const v2f*)(bp + 32 * SX + d0);
            v2f b3 = *(const v2f*)(bp + 48 * SX + d0);
            acc0 = __builtin_amdgcn_wmma_f32_16x16x4_f32(false, a, false, b0, (short)0, acc0, false, false);
            acc1 = __builtin_amdgcn_wmma_f32_16x16x4_f32(false, a, false, b1, (short)0, acc1, false, false);
            acc2 = __builtin_amdgcn_wmma_f32_16x16x4_f32(false, a, false, b2, (short)0, acc2, false, false);
            acc3 = __builtin_amdgcn_wmma_f32_16x16x4_f32(false, a, false, b3, (short)0, acc3, false, false);
        }

        // score = 2*cross - e_sq (x_sq row-constant: irrelevant for argmax)
        const int nb  = c * BN;
        const float eq0 = eqs[nb + mrow];
        const float eq1 = eqs[nb + 16 + mrow];
        const float eq2 = eqs[nb + 32 + mrow];
        const float eq3 = eqs[nb + 48 + mrow];
        #pragma unroll
        for (int r = 0; r < 8; ++r) {
            float s0 = 2.0f * acc0[r] - eq0;
            float s1 = 2.0f * acc1[r] - eq1;
            float s2 = 2.0f * acc2[r] - eq2;
            float s3 = 2.0f * acc3[r] - eq3;
            if (s0 > bestVal[r]) { bestVal[r] = s0; bestIdx[r] = nb + mrow; }
            if (s1 > bestVal[r]) { bestVal[r] = s1; bestIdx[r] = nb + 16 + mrow; }
            if (s2 > bestVal[r]) { bestVal[r] = s2; bestIdx[r] = nb + 32 + mrow; }
            if (s3 > bestVal[r]) { bestVal[r] = s3; bestIdx[r] = nb + 48 + mrow; }
        }

        wait_async0();    // own wave's next-chunk loads landed
        __syncthreads();  // everyone done reading current buffer; next buffer visible
    }

    // per-wave argmax across the 16 lanes of each half (rows differ per half)
    #pragma unroll
    for (int r = 0; r < 8; ++r) {
        float v = bestVal[r];
        int   i = bestIdx[r];
        #pragma unroll
        for (int m = 1; m <= 8; m <<= 1) {
            float ov = __shfl_xor(v, m, 32);
            int   oi = __shfl_xor(i, m, 32);
            if (ov > v || (ov == v && oi < i)) { v = ov; i = oi; }
        }
        if (mrow == 0) {
            int rowLocal = w * 16 + r + hi * 8;
            finalIdx[rowLocal] = i;
            out_ind[rowBlock + rowLocal] = (float)i;
            atomicAdd(&ws_counts[i], 1.0f);
        }
    }
    __syncthreads();

    // gather quantize = embed[idx] and EMA scatter (2 threads per row, 64 floats each)
    {
        int r    = t >> 1;
        int part = t & 1;
        int idx  = finalIdx[r];
        const float* ep = embed     + (size_t)idx * DIM + part * 64;
        float*       qp = out_quant + (size_t)(rowBlock + r) * DIM + part * 64;
        float*       sp = ws_sums   + (size_t)idx * DIM + part * 64;
        const float* xp = &xs[r * SX + part * 64];
        #pragma unroll
        for (int j = 0; j < 64; j += 4) {
            *(v4f*)(qp + j) = *(const v4f*)(ep + j);
            atomicAdd(sp + j + 0, xp[j + 0]);
            atomicAdd(sp + j + 1, xp[j + 1]);
            atomicAdd(sp + j + 2, xp[j + 2]);
            atomicAdd(sp + j + 3, xp[j + 3]);
        }
    }
}

// ---------------------------------------------------------------- finalize: cluster size + total
__global__ __launch_bounds__(1024)
void vq_cluster(const float* __restrict__ cs_in, const float* __restrict__ ws_counts,
                float* __restrict__ out_cs, float* __restrict__ ws_total)
{
    __shared__ float red[32];
    int t = threadIdx.x;
    float v = cs_in[t] * 0.99f + 0.01f * ws_counts[t];
    out_cs[t] = v;
    float s = v;
    #pragma unroll
    for (int m = 16; m >= 1; m >>= 1) s += __shfl_xor(s, m, 32);
    if ((t & 31) == 0) red[t >> 5] = s;
    __syncthreads();
    if (t < 32) {
        float s2 = red[t];
        #pragma unroll
        for (int m = 16; m >= 1; m >>= 1) s2 += __shfl_xor(s2, m, 32);
        if (t == 0) ws_total[0] = s2;
    }
}

// ---------------------------------------------------------------- finalize: embed_avg + new embed
__global__ __launch_bounds__(256)
void vq_embed(const float* __restrict__ ea_in, const float* __restrict__ ws_sums,
              const float* __restrict__ out_cs, const float* __restrict__ ws_total,
              float* __restrict__ out_ea, float* __restrict__ out_embed)
{
    int i = blockIdx.x * 256 + threadIdx.x;
    int k = i >> 7;
    float nea = ea_in[i] * 0.99f + 0.01f * ws_sums[i];
    out_ea[i] = nea;
    float total    = ws_total[0];
    float ncs      = out_cs[k];
    float smoothed = (ncs + 1e-5f) / (total + (float)KCODES * 1e-5f) * total;
    out_embed[i] = nea / smoothed;
}

// ----------------------------------------------------------------
extern "C" void kernel_launch(void* const* d_in, const int* in_sizes, int n_in,
                              void* d_out, int out_size, void* d_ws, size_t ws_size,
                              hipStream_t stream) {
    const float* x     = (const float*)d_in[0];   // (32,4096,128)
    const float* embed = (const float*)d_in[1];   // (1024,128)
    const float* cs_in = (const float*)d_in[2];   // (1024,)
    const float* ea_in = (const float*)d_in[3];   // (1024,128)

    float* out = (float*)d_out;
    float* out_quant = out;                                   // 16,777,216
    float* out_ind   = out_quant + (size_t)NROWS * DIM;       //    131,072
    float* out_cs    = out_ind + NROWS;                       //      1,024
    float* out_ea    = out_cs + KCODES;                       //    131,072
    float* out_embed = out_ea + (size_t)KCODES * DIM;         //    131,072

    float* ws        = (float*)d_ws;
    float* ws_counts = ws;                                    // 1024
    float* ws_sums   = ws_counts + KCODES;                    // 131072
    float* ws_total  = ws_sums + (size_t)KCODES * DIM;        // 1
    float* ws_esq    = ws_total + 4;                          // 1024

    const int nzero = KCODES + KCODES * DIM + 1;
    vq_zero<<<(nzero + 255) / 256, 256, 0, stream>>>(ws, nzero);

    vq_esq<<<(KCODES * 32 + 255) / 256, 256, 0, stream>>>(embed, ws_esq);

    vq_main<<<NROWS / BM, 256, 0, stream>>>(x, embed, ws_esq,
                                            out_quant, out_ind, ws_counts, ws_sums);

    vq_cluster<<<1, 1024, 0, stream>>>(cs_in, ws_counts, out_cs, ws_total);

    vq_embed<<<(KCODES * DIM) / 256, 256, 0, stream>>>(ea_in, ws_sums, out_cs,
                                                       ws_total, out_ea, out_embed);
}